// SpatialGather_Module_52905407152962
// MI455X (gfx1250) — compile-verified
//
#include <hip/hip_runtime.h>
#include <hip/hip_bf16.h>

typedef float v2f __attribute__((ext_vector_type(2)));
typedef float v8f __attribute__((ext_vector_type(8)));
typedef int   v4i __attribute__((ext_vector_type(4)));

#define B_BATCH 16
#define K_CLS   19
#define C_CH    512
#define S_SPA   16384          // 128*128
#define SPLIT   8              // s-dim split per (b, ctile)
#define S_CHUNK (S_SPA / SPLIT)   // 2048 spatial per wave
#define LDS_STRIDE 36          // padded dword row stride (16B aligned, bank-conflict free)

// ---- CDNA5 async global->LDS path (guarded; fallback = sync staging) -------
#if defined(__has_builtin)
#  if __has_builtin(__builtin_amdgcn_global_load_async_to_lds_b128)
#    define SG_ASYNC 1
#  endif
#endif

#if defined(SG_ASYNC)
typedef v4i __attribute__((address_space(1)))* sg_g4p;   // global int4*
typedef v4i __attribute__((address_space(3)))* sg_l4p;   // LDS    int4*

__device__ __forceinline__ void sg_async_cp16(const float* g, float* l) {
    __builtin_amdgcn_global_load_async_to_lds_b128(
        (sg_g4p)(void*)g, (sg_l4p)(void*)l, 0, 0);
}
#  if __has_builtin(__builtin_amdgcn_s_wait_asynccnt)
#    define SG_WAIT_ASYNC(n) __builtin_amdgcn_s_wait_asynccnt(n)
#  else
#    define SG_WAIT_ASYNC(n) asm volatile("s_wait_asynccnt %0" :: "n"(n) : "memory")
#  endif
#endif

// ---------------------------------------------------------------------------
// Kernel 1: per-(b,k) row softmax stats: max and 1/sum(exp)
// grid = 304 blocks, 256 threads
// ---------------------------------------------------------------------------
__global__ __launch_bounds__(256)
void sg_softmax_stats(const float* __restrict__ probs, float* __restrict__ stats) {
    const int row = blockIdx.x;               // b*19 + k
    const float* x = probs + (size_t)row * S_SPA;
    __shared__ float red[256];
    const int t = threadIdx.x;

    float m = -3.402823466e38f;
    for (int i = t; i < S_SPA; i += 256) m = fmaxf(m, x[i]);
    red[t] = m; __syncthreads();
    for (int off = 128; off > 0; off >>= 1) {
        if (t < off) red[t] = fmaxf(red[t], red[t + off]);
        __syncthreads();
    }
    m = red[0]; __syncthreads();

    float s = 0.0f;
    for (int i = t; i < S_SPA; i += 256) s += __expf(x[i] - m);
    red[t] = s; __syncthreads();
    for (int off = 128; off > 0; off >>= 1) {
        if (t < off) red[t] += red[t + off];
        __syncthreads();
    }
    if (t == 0) {
        stats[row * 2 + 0] = m;
        stats[row * 2 + 1] = 1.0f / red[0];
    }
}

// ---------------------------------------------------------------------------
// Kernel 2: softmax apply + transpose to p_t[b][s][32] (k padded to 32, zeros)
// grid = (64, 16), 256 threads; each block handles 256 spatial positions
// ---------------------------------------------------------------------------
__global__ __launch_bounds__(256)
void sg_softmax_transpose(const float* __restrict__ probs,
                          const float* __restrict__ stats,
                          float* __restrict__ p_t) {
    __shared__ float tile[256 * 33];          // [s_local][k], padded stride 33
    const int t  = threadIdx.x;
    const int b  = blockIdx.y;
    const int s0 = blockIdx.x * 256;

    #pragma unroll 1
    for (int k = 0; k < 32; ++k) {
        float v = 0.0f;
        if (k < K_CLS) {
            const int row = b * K_CLS + k;
            const float x = probs[(size_t)row * S_SPA + s0 + t];
            v = __expf(x - stats[row * 2]) * stats[row * 2 + 1];
        }
        tile[t * 33 + k] = v;
    }
    __syncthreads();

    // write 256*32 floats contiguously, coalesced float4 stores
    float4* outp = (float4*)(p_t + ((size_t)b * S_SPA + s0) * 32);
    #pragma unroll
    for (int j = 0; j < 8; ++j) {
        const int f4 = t + 256 * j;
        const int s  = f4 >> 3;               // 8 float4 per 32-wide row
        const int k0 = (f4 & 7) * 4;
        float4 v;
        v.x = tile[s * 33 + k0 + 0];
        v.y = tile[s * 33 + k0 + 1];
        v.z = tile[s * 33 + k0 + 2];
        v.w = tile[s * 33 + k0 + 3];
        outp[f4] = v;
    }
}

// ---------------------------------------------------------------------------
// Kernel 3: WMMA GEMM.  One wave (32 threads) per (b, ctile, s-split).
//   D[16c x 16k] += A[16c x 4s] * B[4s x 16k]  via v_wmma_f32_16x16x4_f32
// A streamed into LDS with async global->LDS b128 ops, double-buffered;
// B read directly from L2-resident p_t with the ISA B-fragment layout.
// grid = 16*32*8 = 4096 blocks of 32 threads
// ---------------------------------------------------------------------------
__global__ __launch_bounds__(32)
void sg_wmma_gemm(const float* __restrict__ feats,
                  const float* __restrict__ p_t,
                  float* __restrict__ partials) {
    __shared__ float As[2][16 * LDS_STRIDE];

    const int lane = threadIdx.x;
    const int idx  = blockIdx.x;              // (b<<8) | (ct<<3) | sc
    const int sc   = idx & (SPLIT - 1);
    const int ct   = (idx >> 3) & 31;
    const int b    = idx >> 8;

    const size_t s_base = (size_t)sc * S_CHUNK;
    const float* fbase  = feats + ((size_t)b * C_CH + (size_t)ct * 16) * S_SPA;
    const float* pbase  = p_t + (size_t)b * S_SPA * 32;

    // A staging map: 4 ops, each covers 4 rows x 32 cols (16B per lane)
    const int arow = lane >> 3;               // 0..3
    const int acol = (lane & 7) * 4;          // 0..28
    // A fragment map (ISA 16x4 f32): lane<16 -> M=lane, K=ss..ss+1
    //                                lane>=16 -> M=lane-16, K=ss+2..ss+3
    const int a_m    = lane & 15;
    const int a_koff = (lane >> 4) * 2;
    // B fragment map (ISA 4x16): VGPR0 = rows K0(l0-15)/K2(l16-31), VGPR1 = K1/K3
    const int b_col    = lane & 15;
    const int b_rowoff = (lane >> 4) * 2;

    v8f acc0 = {};                            // N-tile 0: k = 0..15
    v8f acc1 = {};                            // N-tile 1: k = 16..31 (19..31 zero)

#if defined(SG_ASYNC)
    // prefetch chunk 0 into buffer 0
    {
        #pragma unroll
        for (int j = 0; j < 4; ++j) {
            const int row = j * 4 + arow;
            sg_async_cp16(fbase + (size_t)row * S_SPA + s_base + acol,
                          &As[0][row * LDS_STRIDE + acol]);
        }
    }
    for (int s32 = 0; s32 < S_CHUNK; s32 += 32) {
        const int cur = (s32 >> 5) & 1;
        // prefetch next chunk (wraps on last iter; harmless, never read)
        const size_t s_nxt = s_base + ((s32 + 32) & (S_CHUNK - 1));
        #pragma unroll
        for (int j = 0; j < 4; ++j) {
            const int row = j * 4 + arow;
            sg_async_cp16(fbase + (size_t)row * S_SPA + s_nxt + acol,
                          &As[cur ^ 1][row * LDS_STRIDE + acol]);
        }
        // async loads complete in order: <=4 outstanding => chunk `cur` resident
        SG_WAIT_ASYNC(4);
        asm volatile("" ::: "memory");

        const float* As_c = As[cur];
        const size_t s0 = s_base + s32;
        const float* pb = pbase + s0 * 32 + b_rowoff * 32 + b_col;
        #pragma unroll
        for (int ss = 0; ss < 32; ss += 4) {
            const float2 av = *(const float2*)(As_c + a_m * LDS_STRIDE + a_koff + ss);
            v2f a;  a.x = av.x;  a.y = av.y;
            v2f b0, b1;
            b0.x = pb[ss * 32];            b0.y = pb[(ss + 1) * 32];
            b1.x = pb[ss * 32 + 16];       b1.y = pb[(ss + 1) * 32 + 16];
            acc0 = __builtin_amdgcn_wmma_f32_16x16x4_f32(
                       false, a, false, b0, (short)0, acc0, false, false);
            acc1 = __builtin_amdgcn_wmma_f32_16x16x4_f32(
                       false, a, false, b1, (short)0, acc1, false, false);
        }
    }
#else
    for (int s32 = 0; s32 < S_CHUNK; s32 += 32) {
        const size_t s0 = s_base + s32;

        __syncthreads();
        #pragma unroll
        for (int j = 0; j < 4; ++j) {
            const int row = j * 4 + arow;
            const float4 v = *(const float4*)(fbase + (size_t)row * S_SPA + s0 + acol);
            *(float4*)(&As[0][row * LDS_STRIDE + acol]) = v;
        }
        __syncthreads();

        const float* pb = pbase + s0 * 32 + b_rowoff * 32 + b_col;
        #pragma unroll
        for (int ss = 0; ss < 32; ss += 4) {
            const float2 av = *(const float2*)(&As[0][a_m * LDS_STRIDE + a_koff + ss]);
            v2f a;  a.x = av.x;  a.y = av.y;
            v2f b0, b1;
            b0.x = pb[ss * 32];            b0.y = pb[(ss + 1) * 32];
            b1.x = pb[ss * 32 + 16];       b1.y = pb[(ss + 1) * 32 + 16];
            acc0 = __builtin_amdgcn_wmma_f32_16x16x4_f32(
                       false, a, false, b0, (short)0, acc0, false, false);
            acc1 = __builtin_amdgcn_wmma_f32_16x16x4_f32(
                       false, a, false, b1, (short)0, acc1, false, false);
        }
    }
#endif

    // store both 16x16 partial D tiles (raw WMMA layout), 2 KB contiguous
    float* pp = partials + (size_t)idx * 2 * 256;
    *(v8f*)(pp + lane * 8)        = acc0;
    *(v8f*)(pp + 256 + lane * 8)  = acc1;
}

// ---------------------------------------------------------------------------
// Kernel 4: deterministic reduction over the SPLIT partials, undoing the
// WMMA C/D layout: element (m,n) lives at vgpr=m%8, lane=n+16*(m/8).
// out[b,c,k], 155648 elements
// ---------------------------------------------------------------------------
__global__ __launch_bounds__(256)
void sg_reduce(const float* __restrict__ partials, float* __restrict__ out) {
    const int i = blockIdx.x * 256 + threadIdx.x;
    if (i >= B_BATCH * C_CH * K_CLS) return;
    const int k = i % K_CLS;
    const int c = (i / K_CLS) % C_CH;
    const int b = i / (K_CLS * C_CH);
    const int ct = c >> 4, m = c & 15;
    const int nt = k >> 4, n = k & 15;
    const int lane = n + ((m >> 3) << 4);
    const int r    = m & 7;
    const size_t base = ((size_t)(((b << 8) | (ct << 3)) * 2 + nt)) * 256
                        + (size_t)lane * 8 + r;
    float s = 0.0f;
    #pragma unroll
    for (int sc = 0; sc < SPLIT; ++sc) s += partials[base + (size_t)sc * 512];
    out[i] = s;
}

// ---------------------------------------------------------------------------
extern "C" void kernel_launch(void* const* d_in, const int* in_sizes, int n_in,
                              void* d_out, int out_size, void* d_ws, size_t ws_size,
                              hipStream_t stream) {
    const float* feats = (const float*)d_in[0];   // [16,512,128,128] f32
    const float* probs = (const float*)d_in[1];   // [16,19,128,128]  f32
    float* out = (float*)d_out;                   // [16,512,19]      f32

    float* p_t      = (float*)d_ws;                               // 16*16384*32 f32
    float* stats    = p_t + (size_t)B_BATCH * S_SPA * 32;         // 304*2 f32 (pad 1024)
    float* partials = stats + 1024;                               // 4096*2*256 f32

    sg_softmax_stats<<<B_BATCH * K_CLS, 256, 0, stream>>>(probs, stats);
    sg_softmax_transpose<<<dim3(S_SPA / 256, B_BATCH), 256, 0, stream>>>(probs, stats, p_t);
    sg_wmma_gemm<<<B_BATCH * 32 * SPLIT, 32, 0, stream>>>(feats, p_t, partials);
    const int n_out = B_BATCH * C_CH * K_CLS;
    sg_reduce<<<(n_out + 255) / 256, 256, 0, stream>>>(partials, out);
}